// RNN_65292092834117
// MI455X (gfx1250) — compile-verified
//
#include <hip/hip_runtime.h>
#include <stdint.h>

typedef __attribute__((ext_vector_type(2))) float v2f;
typedef __attribute__((ext_vector_type(8))) float v8f;
typedef __attribute__((ext_vector_type(4))) unsigned int v4u;
typedef __attribute__((ext_vector_type(8))) int v8i_;
typedef __attribute__((ext_vector_type(4))) int v4i_;

#define BATCH  64
#define SEQ    33      // T_CAP + 1
#define TCAP   32
#define EMBED  256
#define HIDDEN 512
#define GATES  2048    // 4 * HIDDEN
#define VOCAB  32000
#define MROWS  (BATCH * SEQ)   // 2112

// logits-GEMM tiling
#define KC    32               // K chunk per TDM transfer (dwords per row)
#define NR    128              // rows of W_out per block (N tile)
#define LDSW  36               // LDS row stride in dwords (KC + 4 pad dwords)

// ---------------------------------------------------------------------------
// fp32 WMMA 16x16x4 helpers.
// A from row-major X[M,K]:  a[r] = X[m][k + 2*(lane>>4) + r],  m = base+(lane&15)
// B from row-major W[N,K]:  b[r] = W[n][k + 2*(lane>>4) + r],  n = base+(lane&15)
// Both are one 8-byte load per lane.
// ---------------------------------------------------------------------------
__device__ __forceinline__ v2f frag_ld(const float* __restrict__ p, int row_base,
                                       int ld, int k) {
    int lane = threadIdx.x & 31;
    const float* q = p + (size_t)(row_base + (lane & 15)) * ld
                       + (k + ((lane >> 4) << 1));
    return *(const v2f*)q;
}

__device__ __forceinline__ v8f wmma_f32(v2f a, v2f b, v8f c) {
    return __builtin_amdgcn_wmma_f32_16x16x4_f32(false, a, false, b,
                                                 (short)0, c, false, false);
}

__device__ __forceinline__ float sigmoidf_(float x) {
    return 1.0f / (1.0f + __expf(-x));
}

// ---------------------------------------------------------------------------
// TDM: DMA a [NR rows x KC cols] fp32 tile of row-major W[N=VOCAB, K=HIDDEN]
// into LDS with 4-dword padding per 32-dword row (stride LDSW dwords).
// D# packing per CDNA5 ISA ch.10.8 (group0: count/lds/global/type,
// group1: data_size, pad, tensor dims/strides, tile dims).
// ---------------------------------------------------------------------------
__device__ __forceinline__ void tdm_load_b_tile(const float* gsrc,
                                                unsigned lds_byte_addr) {
    unsigned long long ga = (unsigned long long)(uintptr_t)gsrc;
    v4u g0;
    g0.x = 1u;                                         // count = 1 valid D#
    g0.y = lds_byte_addr;                              // lds_addr
    g0.z = (unsigned)(ga & 0xffffffffu);               // global_addr[31:0]
    g0.w = (unsigned)((ga >> 32) & 0x01ffffffu)        // global_addr[56:32]
         | (2u << 30);                                 // type = 2 (image)
    v8i_ g1;
    g1[0] = (int)((2u << 16)        // data_size = 4 bytes
                | (1u << 20)        // pad_enable
                | (4u << 22)        // pad_interval: 32 dwords
                | (3u << 25));      // pad_amount:   4 dwords
    g1[1] = (int)(((unsigned)HIDDEN & 0xffffu) << 16); // tensor_dim0 = 512
    g1[2] = (int)(((unsigned)VOCAB & 0xffffu) << 16);  // tensor_dim1 = 32000
    g1[3] = (int)((unsigned)KC << 16);                 // tile_dim0 = 32
    g1[4] = NR;                                        // tile_dim1 = 128
    g1[5] = HIDDEN;                                    // tensor_dim0_stride
    g1[6] = 0;
    g1[7] = 0;
    v4i_ z4 = {0, 0, 0, 0};
#if __clang_major__ >= 23
    v8i_ z8 = {0, 0, 0, 0, 0, 0, 0, 0};
    __builtin_amdgcn_tensor_load_to_lds(g0, g1, z4, z4, z8, 0);
#else
    __builtin_amdgcn_tensor_load_to_lds(g0, g1, z4, z4, 0);
#endif
}

// ---------------------------------------------------------------------------
// Zero-init c-state and h0
// ---------------------------------------------------------------------------
__global__ void __launch_bounds__(256) zero_kernel(float* __restrict__ p, int n4) {
    int i = blockIdx.x * blockDim.x + threadIdx.x;
    if (i < n4) ((float4*)p)[i] = make_float4(0.f, 0.f, 0.f, 0.f);
}

// ---------------------------------------------------------------------------
// Build x[b][s][e]: s==0 -> features[b], else embedding[captions[b][s-1]]
// ---------------------------------------------------------------------------
__global__ void __launch_bounds__(64) build_x_kernel(
        const float* __restrict__ features, const int* __restrict__ captions,
        const float* __restrict__ embedding, float* __restrict__ X) {
    int row = blockIdx.x;            // b*SEQ + s
    int b = row / SEQ, s = row - b * SEQ;
    const float* src;
    if (s == 0) src = features + (size_t)b * EMBED;
    else        src = embedding + (size_t)captions[b * TCAP + (s - 1)] * EMBED;
    float4 v = ((const float4*)src)[threadIdx.x];
    ((float4*)(X + (size_t)row * EMBED))[threadIdx.x] = v;
}

// ---------------------------------------------------------------------------
// Xp[2112,2048] = X[2112,256] @ W_ih^T + (b_ih + b_hh)
// block = 256 (8 waves); wave -> 16x32 tile; grid (132, 8)
// ---------------------------------------------------------------------------
__global__ void __launch_bounds__(256) xproj_kernel(
        const float* __restrict__ X, const float* __restrict__ W,
        const float* __restrict__ b_ih, const float* __restrict__ b_hh,
        float* __restrict__ Xp) {
    int wave = threadIdx.x >> 5;
    int bm = blockIdx.x * 16;
    int bn = blockIdx.y * 256 + wave * 32;
    v8f c0 = {}, c1 = {};
    for (int k = 0; k < EMBED; k += 4) {
        v2f a  = frag_ld(X, bm, EMBED, k);
        v2f b0 = frag_ld(W, bn, EMBED, k);
        v2f b1 = frag_ld(W, bn + 16, EMBED, k);
        c0 = wmma_f32(a, b0, c0);
        c1 = wmma_f32(a, b1, c1);
    }
    int lane = threadIdx.x & 31;
    int n0 = bn + (lane & 15), n1 = n0 + 16;
    int m0 = bm + ((lane >> 4) << 3);
    float bias0 = b_ih[n0] + b_hh[n0];
    float bias1 = b_ih[n1] + b_hh[n1];
#pragma unroll
    for (int r = 0; r < 8; ++r) {
        Xp[(size_t)(m0 + r) * GATES + n0] = c0[r] + bias0;
        Xp[(size_t)(m0 + r) * GATES + n1] = c1[r] + bias1;
    }
}

// ---------------------------------------------------------------------------
// One LSTM timestep (fused gate-GEMM + cell).
// block = 128 (4 waves = gates i,f,g,o); block tile = 16(batch) x 16(hidden).
// hs laid out [b][s][h] so GEMM-M order == logits row order (b*SEQ+s).
// ---------------------------------------------------------------------------
__global__ void __launch_bounds__(128) lstm_step_kernel(
        const float* __restrict__ Xp, const float* __restrict__ W_hh,
        const float* __restrict__ h_prev, int h_prev_ld,
        float* __restrict__ hs_t,      // = hs + t*HIDDEN, row stride SEQ*HIDDEN
        float* __restrict__ c_state, int t) {
    __shared__ float lds_g[4 * 256];
    int wave = threadIdx.x >> 5;
    int lane = threadIdx.x & 31;
    int bm = blockIdx.x * 16;          // batch tile
    int bn = blockIdx.y * 16;          // hidden tile (within 512)
    int gcol = wave * HIDDEN + bn + (lane & 15);

    v8f acc;
    int m0 = ((lane >> 4) << 3);
#pragma unroll
    for (int r = 0; r < 8; ++r)
        acc[r] = Xp[((size_t)(bm + m0 + r) * SEQ + t) * GATES + gcol];

    for (int k = 0; k < HIDDEN; k += 4) {
        v2f a = frag_ld(h_prev, bm, h_prev_ld, k);
        v2f b = frag_ld(W_hh, wave * HIDDEN + bn, HIDDEN, k);
        acc = wmma_f32(a, b, acc);
    }

#pragma unroll
    for (int r = 0; r < 8; ++r)
        lds_g[wave * 256 + lane * 8 + r] = acc[r];
    __syncthreads();

    for (int e = threadIdx.x; e < 256; e += 128) {
        int m = e >> 4, n = e & 15;
        int sl = (((m >> 3) << 4) | n) * 8 + (m & 7);
        float gi = lds_g[0 * 256 + sl];
        float gf = lds_g[1 * 256 + sl];
        float gg = lds_g[2 * 256 + sl];
        float go = lds_g[3 * 256 + sl];
        int row = bm + m, col = bn + n;
        float cp = c_state[(size_t)row * HIDDEN + col];
        float cn = sigmoidf_(gf) * cp + sigmoidf_(gi) * tanhf(gg);
        float hn = sigmoidf_(go) * tanhf(cn);
        c_state[(size_t)row * HIDDEN + col] = cn;
        hs_t[(size_t)row * (SEQ * HIDDEN) + col] = hn;
    }
}

// ---------------------------------------------------------------------------
// logits[2112,32000] = hs[2112,512] @ W_out^T + b_out
// block = 256 (8 waves, 2M x 4N); block tile 64M x 128N; wave tile 32x32.
// W_out tiles (128 x 32) DMA'd to LDS by the Tensor Data Mover, double
// buffered on TENSORcnt. B fragment reads from LDS are bank-conflict free
// (row stride 36 dwords). grid (33, 250).
// ---------------------------------------------------------------------------
__global__ void __launch_bounds__(256) logits_kernel(
        const float* __restrict__ hs, const float* __restrict__ W_out,
        const float* __restrict__ b_out, float* __restrict__ out) {
    __shared__ __align__(16) float bsm[2][NR * LDSW];
    int wave = threadIdx.x >> 5;
    int lane = threadIdx.x & 31;
    int bm  = blockIdx.x * 64;         // M tile base (row = b*SEQ+s)
    int bnb = blockIdx.y * NR;         // N tile base (vocab)
    int wm = bm + (wave >> 2) * 32;    // this wave's 32 M rows
    int wn = (wave & 3) * 32;          // this wave's 32 N cols (local)

    v8f c00 = {}, c01 = {}, c10 = {}, c11 = {};

    if (threadIdx.x == 0)
        tdm_load_b_tile(W_out + (size_t)bnb * HIDDEN,
                        (unsigned)(uintptr_t)(void*)&bsm[0][0]);

    const int NCH = HIDDEN / KC;       // 16 K-chunks
    for (int c = 0; c < NCH; ++c) {
        if (threadIdx.x == 0) {
            if (c + 1 < NCH) {
                tdm_load_b_tile(W_out + (size_t)bnb * HIDDEN + (c + 1) * KC,
                                (unsigned)(uintptr_t)(void*)&bsm[(c + 1) & 1][0]);
                __builtin_amdgcn_s_wait_tensorcnt(1);   // chunk c landed
            } else {
                __builtin_amdgcn_s_wait_tensorcnt(0);
            }
        }
        __syncthreads();

        const float* bs = &bsm[c & 1][0];
        int kb = c * KC;
        int koffl = ((lane >> 4) << 1);
        int nrow0 = (wn + (lane & 15)) * LDSW;
        int nrow1 = (wn + 16 + (lane & 15)) * LDSW;
#pragma unroll
        for (int kk = 0; kk < KC; kk += 4) {
            v2f a0 = frag_ld(hs, wm, HIDDEN, kb + kk);
            v2f a1 = frag_ld(hs, wm + 16, HIDDEN, kb + kk);
            v2f b0 = *(const v2f*)(bs + nrow0 + kk + koffl);
            v2f b1 = *(const v2f*)(bs + nrow1 + kk + koffl);
            c00 = wmma_f32(a0, b0, c00);
            c01 = wmma_f32(a0, b1, c01);
            c10 = wmma_f32(a1, b0, c10);
            c11 = wmma_f32(a1, b1, c11);
        }
        __syncthreads();               // all reads of bsm[c&1] done
    }

    int n0 = bnb + wn + (lane & 15), n1 = n0 + 16;
    int m0 = wm + ((lane >> 4) << 3);
    float bb0 = b_out[n0], bb1 = b_out[n1];
#pragma unroll
    for (int r = 0; r < 8; ++r) {
        out[(size_t)(m0 + r) * VOCAB + n0]      = c00[r] + bb0;
        out[(size_t)(m0 + r) * VOCAB + n1]      = c01[r] + bb1;
        out[(size_t)(m0 + 16 + r) * VOCAB + n0] = c10[r] + bb0;
        out[(size_t)(m0 + 16 + r) * VOCAB + n1] = c11[r] + bb1;
    }
}

// ---------------------------------------------------------------------------
extern "C" void kernel_launch(void* const* d_in, const int* in_sizes, int n_in,
                              void* d_out, int out_size, void* d_ws, size_t ws_size,
                              hipStream_t stream) {
    const float* features  = (const float*)d_in[0];
    const int*   captions  = (const int*)  d_in[1];
    // d_in[2] = lengths: unused by the reference's logits computation
    const float* embedding = (const float*)d_in[3];
    const float* W_ih      = (const float*)d_in[4];
    const float* W_hh      = (const float*)d_in[5];
    const float* b_ih      = (const float*)d_in[6];
    const float* b_hh      = (const float*)d_in[7];
    const float* W_out     = (const float*)d_in[8];
    const float* b_out     = (const float*)d_in[9];
    float* out = (float*)d_out;

    float* ws  = (float*)d_ws;
    float* X   = ws;                              // 2112 * 256
    float* Xp  = X   + (size_t)MROWS * EMBED;     // 2112 * 2048
    float* hs  = Xp  + (size_t)MROWS * GATES;     // 64 * 33 * 512  ([b][s][h])
    float* cst = hs  + (size_t)BATCH * SEQ * HIDDEN;  // 64 * 512
    float* h0  = cst + (size_t)BATCH * HIDDEN;        // 64 * 512 (zeros)

    zero_kernel<<<dim3(64), 256, 0, stream>>>(cst, 2 * BATCH * HIDDEN / 4);

    build_x_kernel<<<dim3(MROWS), 64, 0, stream>>>(features, captions, embedding, X);

    xproj_kernel<<<dim3(MROWS / 16, GATES / 256), 256, 0, stream>>>(
        X, W_ih, b_ih, b_hh, Xp);

    for (int t = 0; t < SEQ; ++t) {
        const float* hp = (t == 0) ? h0 : (hs + (size_t)(t - 1) * HIDDEN);
        int hp_ld = (t == 0) ? HIDDEN : SEQ * HIDDEN;
        lstm_step_kernel<<<dim3(BATCH / 16, HIDDEN / 16), 128, 0, stream>>>(
            Xp, W_hh, hp, hp_ld, hs + (size_t)t * HIDDEN, cst, t);
    }

    logits_kernel<<<dim3(MROWS / 64, VOCAB / NR), 256, 0, stream>>>(
        hs, W_out, b_out, out);
}